// SampledSoftmaxLossWithLogits_9079560864365
// MI455X (gfx1250) — compile-verified
//
#include <hip/hip_runtime.h>

// ---------------------------------------------------------------------------
// Sampled softmax loss (gather + logsumexp + mean) for MI455X / gfx1250.
//
// Kernel 1: one block per row (4096 rows, 256 threads = 8 wave32).
//   - one global_load_b128 per thread for 4 sample indices (coalesced),
//   - 4 independent random gathers per thread (gather/latency-bound; deep
//     LOADcnt pipelining, ~32 waves of gathers in flight per WGP),
//   - values stay in registers for both logsumexp passes,
//   - stable logsumexp: wave32 shuffle-tree + LDS cross-wave reductions,
//   - writes per-row loss to d_ws (deterministic, no float atomics).
// Kernel 2: single wave32; deterministic mean of 4096 row losses using
//   V_WMMA_F32_16X16X4_F32 as a reduction engine:
//   A = ones(16x4)  =>  D[m][n] = sum_k B[k][n]; summing D's first VGPR over
//   all 32 lanes gives 2 * (sum of every element placed in B), regardless of
//   the exact 4x16 B lane layout. B carries one partial sum per lane.
// ---------------------------------------------------------------------------

#define BATCH       4096
#define NUM_CLASS   50257
#define NUM_SAMPLED 1024
#define THREADS     256
#define PER_THREAD  (NUM_SAMPLED / THREADS)   // 4
#define NWAVES      (THREADS / 32)            // 8

typedef __attribute__((ext_vector_type(2))) float v2f;
typedef __attribute__((ext_vector_type(8))) float v8f;

__global__ __launch_bounds__(THREADS)
void sampled_softmax_row_kernel(const float* __restrict__ logits,
                                const int*   __restrict__ labels,
                                const int*   __restrict__ sampled_idx,
                                float*       __restrict__ row_loss)
{
    const int row  = blockIdx.x;
    const int t    = threadIdx.x;
    const int lane = t & 31;
    const int wave = t >> 5;

    const float* __restrict__ lrow = logits + (size_t)row * NUM_CLASS;
    const int*   __restrict__ srow = sampled_idx + (size_t)row * NUM_SAMPLED;

    __shared__ float s_true;
    __shared__ float s_max[NWAVES];
    __shared__ float s_sum[NWAVES];

    // Hint the row into cache early (gfx1250: global_prefetch_b8).
    __builtin_prefetch(lrow, 0, 1);

    if (t == 0) {
        s_true = lrow[labels[row]];
    }

    // --- one b128 load for 4 indices, then 4 independent gathers ---
    const int4 ixv = reinterpret_cast<const int4*>(srow)[t];
    int ix[PER_THREAD] = { ixv.x, ixv.y, ixv.z, ixv.w };

    float v[PER_THREAD];
#pragma unroll
    for (int k = 0; k < PER_THREAD; ++k)
        v[k] = lrow[ix[k]];

    __syncthreads();
    const float tl = s_true;

    // --- pass 1: row max over {true, 1024 sampled} ---
    float m = tl;
#pragma unroll
    for (int k = 0; k < PER_THREAD; ++k)
        m = fmaxf(m, v[k]);
#pragma unroll
    for (int off = 16; off > 0; off >>= 1)
        m = fmaxf(m, __shfl_xor(m, off, 32));
    if (lane == 0) s_max[wave] = m;
    __syncthreads();

    float M = s_max[0];
#pragma unroll
    for (int w = 1; w < NWAVES; ++w)
        M = fmaxf(M, s_max[w]);

    // --- pass 2: sum of exp(x - M); values still live in registers ---
    float s = 0.0f;
#pragma unroll
    for (int k = 0; k < PER_THREAD; ++k)
        s += __expf(v[k] - M);
    if (t == 0)
        s += __expf(tl - M);
#pragma unroll
    for (int off = 16; off > 0; off >>= 1)
        s += __shfl_xor(s, off, 32);
    if (lane == 0) s_sum[wave] = s;
    __syncthreads();

    if (t == 0) {
        float tot = s_sum[0];
#pragma unroll
        for (int w = 1; w < NWAVES; ++w)
            tot += s_sum[w];
        const float logz = M + __logf(tot);
        row_loss[row] = logz - tl;
    }
}

// Single-wave deterministic mean via WMMA (EXEC all ones, no divergence
// before the matrix op).
__global__ __launch_bounds__(32)
void reduce_mean_wmma_kernel(const float* __restrict__ row_loss,
                             float*       __restrict__ out)
{
    const int lane = threadIdx.x;   // 0..31, all active

    // 32 coalesced float4 loads per lane (uniform trip count -> EXEC = all 1s).
    const float4* __restrict__ rl4 = (const float4*)row_loss;
    float p = 0.0f;
    for (int i = lane; i < BATCH / 4; i += 32) {
        const float4 x = rl4[i];
        p += (x.x + x.y) + (x.z + x.w);
    }

    v2f a; a[0] = 1.0f; a[1] = 1.0f;   // A = ones(16x4)
    v2f b; b[0] = p;    b[1] = 0.0f;   // B carries one partial per lane
    v8f c = {};
    // D[m][n] = sum_k B[k][n]  (v_wmma_f32_16x16x4_f32)
    c = __builtin_amdgcn_wmma_f32_16x16x4_f32(
        /*neg_a=*/false, a, /*neg_b=*/false, b,
        /*c_mod=*/(short)0, c, /*reuse_a=*/false, /*reuse_b=*/false);

    // c[0]: lanes 0-15 = row M=0 (N=lane), lanes 16-31 = row M=8.
    // Sum over all 32 lanes = 2 * (total of all B elements) = 2 * sum(p).
    float d0 = c[0];
#pragma unroll
    for (int off = 16; off > 0; off >>= 1)
        d0 += __shfl_xor(d0, off, 32);

    if (lane == 0)
        out[0] = d0 * (1.0f / (2.0f * (float)BATCH));
}

extern "C" void kernel_launch(void* const* d_in, const int* in_sizes, int n_in,
                              void* d_out, int out_size, void* d_ws, size_t ws_size,
                              hipStream_t stream)
{
    const float* logits      = (const float*)d_in[0];
    const int*   labels      = (const int*)d_in[1];
    const int*   sampled_idx = (const int*)d_in[2];
    float*       out         = (float*)d_out;
    float*       row_loss    = (float*)d_ws;   // BATCH floats (16 KB)

    sampled_softmax_row_kernel<<<BATCH, THREADS, 0, stream>>>(
        logits, labels, sampled_idx, row_loss);
    reduce_mean_wmma_kernel<<<1, 32, 0, stream>>>(row_loss, out);
}